// pyg_GCN_10282151707718
// MI455X (gfx1250) — compile-verified
//
#include <hip/hip_runtime.h>

typedef __attribute__((ext_vector_type(2))) float v2f;
typedef __attribute__((ext_vector_type(8))) float v8f;

#define N_NODES 50000
#define N_EDGES 800000
#define K_DIM   128

// ---------------------------------------------------------------------------
// Degree / normalization
// ---------------------------------------------------------------------------
__global__ void deg_init_kernel(float* deg, int n) {
    int i = blockIdx.x * blockDim.x + threadIdx.x;
    if (i < n) deg[i] = 1.0f;               // self-loop contributes 1
}

__global__ void deg_count_kernel(const int* __restrict__ dst, float* deg, int e) {
    int i = blockIdx.x * blockDim.x + threadIdx.x;
    if (i < e) unsafeAtomicAdd(&deg[dst[i]], 1.0f);
}

__global__ void deg_finalize_kernel(float* deg, int n) {
    int i = blockIdx.x * blockDim.x + threadIdx.x;
    if (i < n) deg[i] = rsqrtf(deg[i]);     // deg >= 1 always (self loop)
}

// ---------------------------------------------------------------------------
// Dense GEMM  H[N x F] = X[N x 128] @ W[128 x F]  via V_WMMA_F32_16X16X4_F32.
// One wave computes one 16x16 tile of H; wave w of the block handles column
// tile w. blockDim.x = 32*COLTILES, gridDim.x = N/16 (= 3125, exact).
// Optional fused ReLU on the X load (previous layer's activation).
// ---------------------------------------------------------------------------
template <int COLTILES>
__global__ void gcn_gemm_wmma_kernel(const float* __restrict__ X,
                                     const float* __restrict__ W,
                                     float* __restrict__ H,
                                     int relu_in) {
    constexpr int F = 16 * COLTILES;
    __shared__ float xs[16 * 129];          // 129-stride: bank-conflict-free

    const int row0 = blockIdx.x << 4;
    const int tid  = threadIdx.x;

    // Stage the 16 x 128 X tile (coalesced), fusing ReLU of prior layer.
    for (int idx = tid; idx < 16 * K_DIM; idx += 32 * COLTILES) {
        int r = idx >> 7, c = idx & (K_DIM - 1);
        float v = X[(size_t)(row0 + r) * K_DIM + c];
        if (relu_in) v = fmaxf(v, 0.0f);
        xs[r * 129 + c] = v;
    }
    __syncthreads();

    const int wave = tid >> 5;              // column tile
    const int lane = tid & 31;
    const int hi   = lane >> 4;             // 0/1 half-wave
    const int lo   = lane & 15;
    const int c0   = wave << 4;

    v8f acc = {};
    #pragma unroll 4
    for (int k0 = 0; k0 < K_DIM; k0 += 4) {
        const int ka = k0 + 2 * hi;         // ISA layout: K = reg + 2*(lane>>4)
        v2f a, b;
        a.x = xs[lo * 129 + ka];
        a.y = xs[lo * 129 + ka + 1];
        b.x = W[(size_t)ka * F + c0 + lo];
        b.y = W[(size_t)(ka + 1) * F + c0 + lo];
        acc = __builtin_amdgcn_wmma_f32_16x16x4_f32(
            /*neg_a=*/false, a, /*neg_b=*/false, b,
            /*c_mod=*/(short)0, acc, /*reuse_a=*/false, /*reuse_b=*/false);
    }

    // C layout: VGPR v -> row v (lanes 0-15) / row v+8 (lanes 16-31)
    const int rbase = row0 + 8 * hi;
    #pragma unroll
    for (int v = 0; v < 8; ++v)
        H[(size_t)(rbase + v) * F + c0 + lo] = acc[v];
}

// ---------------------------------------------------------------------------
// agg[i,:] = bias + dinv[i]^2 * h[i,:]   (self-loop message + bias, fused)
// ---------------------------------------------------------------------------
__global__ void init_agg_kernel(const float* __restrict__ H,
                                const float* __restrict__ dinv,
                                const float* __restrict__ bias,
                                float* __restrict__ agg,
                                int shift, int total) {
    int idx = blockIdx.x * blockDim.x + threadIdx.x;
    if (idx >= total) return;
    int row = idx >> shift;
    int col = idx & ((1 << shift) - 1);
    float di = dinv[row];
    agg[idx] = bias[col] + di * di * H[idx];
}

// ---------------------------------------------------------------------------
// Edge scatter: one wave per edge; CHUNKS = F/32 coalesced fp32 atomic chunks.
// ---------------------------------------------------------------------------
template <int CHUNKS>
__global__ void edge_scatter_kernel(const float* __restrict__ H,
                                    const int* __restrict__ src,
                                    const int* __restrict__ dst,
                                    const float* __restrict__ dinv,
                                    float* __restrict__ agg,
                                    int nEdges) {
    constexpr int F = CHUNKS * 32;
    int gw   = (blockIdx.x * blockDim.x + threadIdx.x) >> 5;
    int lane = threadIdx.x & 31;
    if (gw >= nEdges) return;
    int s = src[gw], d = dst[gw];
    float nrm = dinv[s] * dinv[d];
    const float* hrow = H + (size_t)s * F;
    float* arow = agg + (size_t)d * F;
    #pragma unroll
    for (int j = 0; j < CHUNKS; ++j)
        unsafeAtomicAdd(&arow[lane + 32 * j], nrm * hrow[lane + 32 * j]);
}

// ---------------------------------------------------------------------------
// Final: out = log_softmax(relu(agg)) over 64 columns. One wave per row.
// ---------------------------------------------------------------------------
__global__ void logsoftmax64_kernel(const float* __restrict__ agg,
                                    float* __restrict__ out, int n) {
    int gw   = (blockIdx.x * blockDim.x + threadIdx.x) >> 5;
    int lane = threadIdx.x & 31;
    if (gw >= n) return;
    const float* row = agg + (size_t)gw * 64;
    float v0 = fmaxf(row[lane], 0.0f);
    float v1 = fmaxf(row[lane + 32], 0.0f);
    float m = fmaxf(v0, v1);
    #pragma unroll
    for (int off = 16; off > 0; off >>= 1)
        m = fmaxf(m, __shfl_xor(m, off, 32));
    float s = expf(v0 - m) + expf(v1 - m);
    #pragma unroll
    for (int off = 16; off > 0; off >>= 1)
        s += __shfl_xor(s, off, 32);
    float l = m + logf(s);
    out[(size_t)gw * 64 + lane]      = v0 - l;
    out[(size_t)gw * 64 + lane + 32] = v1 - l;
}

// ---------------------------------------------------------------------------
extern "C" void kernel_launch(void* const* d_in, const int* in_sizes, int n_in,
                              void* d_out, int out_size, void* d_ws, size_t ws_size,
                              hipStream_t stream) {
    (void)in_sizes; (void)n_in; (void)out_size; (void)ws_size;

    const float* x  = (const float*)d_in[0];
    const int*   ei = (const int*)  d_in[1];
    const float* W1 = (const float*)d_in[2];
    const float* b1 = (const float*)d_in[3];
    const float* W2 = (const float*)d_in[4];
    const float* b2 = (const float*)d_in[5];
    const float* W3 = (const float*)d_in[6];
    const float* b3 = (const float*)d_in[7];
    const int* src = ei;
    const int* dst = ei + N_EDGES;
    float* out = (float*)d_out;

    // Workspace layout: dinv | bufA (N x 128) | bufB (N x 128)
    char* ws = (char*)d_ws;
    float* dinv = (float*)ws;
    size_t off = ((size_t)N_NODES * sizeof(float) + 255) & ~(size_t)255;
    float* bufA = (float*)(ws + off);
    float* bufB = bufA + (size_t)N_NODES * 128;

    const int rowTiles = N_NODES / 16;                 // 3125 (exact)

    // Degrees -> dinv
    deg_init_kernel<<<(N_NODES + 255) / 256, 256, 0, stream>>>(dinv, N_NODES);
    deg_count_kernel<<<(N_EDGES + 255) / 256, 256, 0, stream>>>(dst, dinv, N_EDGES);
    deg_finalize_kernel<<<(N_NODES + 255) / 256, 256, 0, stream>>>(dinv, N_NODES);

    // ---- Layer 1: 128 -> 128 ----
    gcn_gemm_wmma_kernel<8><<<rowTiles, 256, 0, stream>>>(x, W1, bufA, 0);
    init_agg_kernel<<<(N_NODES * 128 + 255) / 256, 256, 0, stream>>>(
        bufA, dinv, b1, bufB, 7, N_NODES * 128);
    edge_scatter_kernel<4><<<(N_EDGES * 32 + 255) / 256, 256, 0, stream>>>(
        bufA, src, dst, dinv, bufB, N_EDGES);

    // ---- Layer 2: 128 -> 128 (ReLU of layer 1 fused into GEMM load) ----
    gcn_gemm_wmma_kernel<8><<<rowTiles, 256, 0, stream>>>(bufB, W2, bufA, 1);
    init_agg_kernel<<<(N_NODES * 128 + 255) / 256, 256, 0, stream>>>(
        bufA, dinv, b2, bufB, 7, N_NODES * 128);
    edge_scatter_kernel<4><<<(N_EDGES * 32 + 255) / 256, 256, 0, stream>>>(
        bufA, src, dst, dinv, bufB, N_EDGES);

    // ---- Layer 3: 128 -> 64 (ReLU of layer 2 fused into GEMM load) ----
    gcn_gemm_wmma_kernel<4><<<rowTiles, 128, 0, stream>>>(bufB, W3, bufA, 1);
    init_agg_kernel<<<(N_NODES * 64 + 255) / 256, 256, 0, stream>>>(
        bufA, dinv, b3, bufB, 6, N_NODES * 64);
    edge_scatter_kernel<2><<<(N_EDGES * 32 + 255) / 256, 256, 0, stream>>>(
        bufA, src, dst, dinv, bufB, N_EDGES);

    // ---- ReLU + log_softmax -> d_out ----
    logsoftmax64_kernel<<<(N_NODES * 32 + 255) / 256, 256, 0, stream>>>(bufB, out, N_NODES);
}